// MultiHeadCoAttention_63385127355034
// MI455X (gfx1250) — compile-verified
//
#include <hip/hip_runtime.h>
#include <hip/hip_bf16.h>
#include <math.h>

// Problem constants
#define BATCH 2
#define CCH   64
#define HWSZ  1024   // 32*32
#define NHEAD 8
#define HDIM  8

typedef __attribute__((ext_vector_type(2))) float v2f;
typedef __attribute__((ext_vector_type(8))) float v8f;

// ---------------------------------------------------------------------------
// D(64 x 1024) = W(64x64) @ X(64 x 1024) + bias  (+ optional residual R)
// One wave (32 lanes) computes one 16x16 tile via V_WMMA_F32_16X16X4_F32.
// Layouts per CDNA5 ISA:
//   A[m][k]: VGPR (k&1), lane (k>>1)*16 + m
//   B[k][n]: VGPR (k&1), lane (k>>1)*16 + n
//   D[m][n]: VGPR r holds rows m=r (lanes 0-15) and m=r+8 (lanes 16-31), n=lane%16
// ---------------------------------------------------------------------------
__global__ __launch_bounds__(32)
void gemm64_wmma(const float* __restrict__ W, const float* __restrict__ bias,
                 const float* __restrict__ X, const float* __restrict__ R,
                 float* __restrict__ D, int addR) {
  const int nt   = blockIdx.x;          // 0..63  (n tile)
  const int mt   = blockIdx.y;          // 0..3   (m tile)
  const int b    = blockIdx.z;          // batch
  const int lane = threadIdx.x;         // 0..31
  const int lm   = lane & 15;
  const int lh   = lane >> 4;           // 0 or 1

  const float* Xb = X + b * CCH * HWSZ;
  const int arow = mt * 16 + lm;        // A row for this lane
  const int col  = nt * 16 + lm;        // output column for this lane

  v8f acc = {};
  for (int k0 = 0; k0 < CCH; k0 += 4) {
    v2f a, bm;
    // A operand: W rows, K-chunk k0 .. k0+3
    a[0]  = W[arow * CCH + k0 + 2 * lh + 0];
    a[1]  = W[arow * CCH + k0 + 2 * lh + 1];
    // B operand: X rows k0..k0+3, columns nt*16..+15
    bm[0] = Xb[(k0 + 2 * lh + 0) * HWSZ + col];
    bm[1] = Xb[(k0 + 2 * lh + 1) * HWSZ + col];
    acc = __builtin_amdgcn_wmma_f32_16x16x4_f32(
        /*neg_a=*/false, a, /*neg_b=*/false, bm,
        /*c_mod=*/(short)0, acc, /*reuse_a=*/false, /*reuse_b=*/false);
  }

  float* Db = D + b * CCH * HWSZ;
  const float* Rb = R + b * CCH * HWSZ;
#pragma unroll
  for (int r = 0; r < 8; ++r) {
    const int row = mt * 16 + r + 8 * lh;
    float v = acc[r] + bias[row];
    if (addR) v += Rb[row * HWSZ + col];
    Db[row * HWSZ + col] = v;
  }
}

// ---------------------------------------------------------------------------
// Collapse head_dim out of Wk/Wv: Wkh[h][c] = sum_n Wk[h*8+n][c], same for bias.
// ---------------------------------------------------------------------------
__global__ void reduce_w(const float* __restrict__ Wk, const float* __restrict__ bk,
                         const float* __restrict__ Wv, const float* __restrict__ bv,
                         float* __restrict__ Wkh, float* __restrict__ bkh,
                         float* __restrict__ Wvh, float* __restrict__ bvh) {
  int t = blockIdx.x * blockDim.x + threadIdx.x;   // 0..511
  if (t >= NHEAD * CCH) return;
  int h = t >> 6;          // /64
  int c = t & 63;
  float sk = 0.f, sv = 0.f;
#pragma unroll
  for (int n = 0; n < HDIM; ++n) {
    sk += Wk[(h * HDIM + n) * CCH + c];
    sv += Wv[(h * HDIM + n) * CCH + c];
  }
  Wkh[t] = sk;
  Wvh[t] = sv;
  if (c == 0) {
    float bs = 0.f, bvs = 0.f;
#pragma unroll
    for (int n = 0; n < HDIM; ++n) { bs += bk[h * HDIM + n]; bvs += bv[h * HDIM + n]; }
    bkh[h] = bs;
    bvh[h] = bvs;
  }
}

// ---------------------------------------------------------------------------
// Ksum[b,h,w] = Wkh[h,:] . x2[b,:,w] + bkh[h]; same for Vsum. Tiny 8-row GEMM.
// ---------------------------------------------------------------------------
__global__ void kv_sum(const float* __restrict__ Wkh, const float* __restrict__ bkh,
                       const float* __restrict__ Wvh, const float* __restrict__ bvh,
                       const float* __restrict__ x2,
                       float* __restrict__ Ksum, float* __restrict__ Vsum) {
  int idx = blockIdx.x * blockDim.x + threadIdx.x;   // 0 .. B*8*1024-1
  int w = idx & (HWSZ - 1);
  int h = (idx >> 10) & (NHEAD - 1);
  int b = idx >> 13;
  const float* Xb = x2 + b * CCH * HWSZ;
  float sk = bkh[h], sv = bvh[h];
#pragma unroll 8
  for (int c = 0; c < CCH; ++c) {
    float xv = Xb[c * HWSZ + w];
    sk = fmaf(Wkh[h * CCH + c], xv, sk);
    sv = fmaf(Wvh[h * CCH + c], xv, sv);
  }
  Ksum[idx] = sk;
  Vsum[idx] = sv;
}

// ---------------------------------------------------------------------------
// Per (b,h): max and min of Ksum over the 1024 spatial positions.
// ---------------------------------------------------------------------------
__global__ void kminmax(const float* __restrict__ Ksum, float* __restrict__ mm) {
  __shared__ float smax[256];
  __shared__ float smin[256];
  const int bh = blockIdx.x;                 // 0..15
  const float* K = Ksum + bh * HWSZ;
  const int t = threadIdx.x;
  float mx = -3.402823466e38f, mn = 3.402823466e38f;
  for (int i = t; i < HWSZ; i += 256) {
    float v = K[i];
    mx = fmaxf(mx, v);
    mn = fminf(mn, v);
  }
  smax[t] = mx; smin[t] = mn;
  __syncthreads();
  for (int s = 128; s > 0; s >>= 1) {
    if (t < s) {
      smax[t] = fmaxf(smax[t], smax[t + s]);
      smin[t] = fminf(smin[t], smin[t + s]);
    }
    __syncthreads();
  }
  if (t == 0) { mm[bh * 2] = smax[0]; mm[bh * 2 + 1] = smin[0]; }
}

// ---------------------------------------------------------------------------
// Attention core. For each scalar s = Q[b,h,n,q]:
//   logits_w = s * Ksum[b,h,w]; out = sum_w softmax(logits)_w * Vsum[b,h,w]
// Work in exp2 domain: t = s*log2(e); stabilizer m2 = (t>=0 ? t*maxK : t*minK),
// which equals log2e * max_w(logits) -> identical stabilized softmax.
// ---------------------------------------------------------------------------
__global__ __launch_bounds__(256)
void attn_kernel(const float* __restrict__ Q, const float* __restrict__ Ksum,
                 const float* __restrict__ Vsum, const float* __restrict__ mm,
                 float* __restrict__ O) {
  __shared__ float sK[HWSZ];
  __shared__ float sV[HWSZ];
  const int bh = blockIdx.z;   // b*8 + h, 0..15
  const int n  = blockIdx.y;   // 0..7
  const int qb = blockIdx.x;   // 0..3
  const int t  = threadIdx.x;  // 0..255

  const float* Kp = Ksum + bh * HWSZ;
  const float* Vp = Vsum + bh * HWSZ;
  for (int i = t; i < HWSZ; i += 256) { sK[i] = Kp[i]; sV[i] = Vp[i]; }
  __syncthreads();

  const int b = bh >> 3, h = bh & 7;
  const int q = qb * 256 + t;
  const int ch = h * HDIM + n;
  const float s = Q[(b * CCH + ch) * HWSZ + q];
  const float L2E = 1.4426950408889634f;
  const float tt = s * L2E;
  const float mx = mm[bh * 2], mn = mm[bh * 2 + 1];
  const float m2 = (tt >= 0.f) ? tt * mx : tt * mn;

  float denom = 0.f, numer = 0.f;
#pragma unroll 4
  for (int w = 0; w < HWSZ; ++w) {
    float e = __builtin_amdgcn_exp2f(fmaf(tt, sK[w], -m2));
    denom += e;
    numer = fmaf(e, sV[w], numer);
  }
  O[(b * CCH + ch) * HWSZ + q] = numer / denom;
}

// ---------------------------------------------------------------------------
extern "C" void kernel_launch(void* const* d_in, const int* in_sizes, int n_in,
                              void* d_out, int out_size, void* d_ws, size_t ws_size,
                              hipStream_t stream) {
  const float* x1 = (const float*)d_in[0];
  const float* x2 = (const float*)d_in[1];
  const float* Wq = (const float*)d_in[2];
  const float* bq = (const float*)d_in[3];
  const float* Wk = (const float*)d_in[4];
  const float* bk = (const float*)d_in[5];
  const float* Wv = (const float*)d_in[6];
  const float* bv = (const float*)d_in[7];
  const float* Wo = (const float*)d_in[8];
  const float* bo = (const float*)d_in[9];
  float* out = (float*)d_out;

  // Workspace carve-up (floats)
  float* ws   = (float*)d_ws;
  float* Qbuf = ws;                         // B*C*HW = 131072
  float* Obuf = Qbuf + BATCH * CCH * HWSZ;  // 131072
  float* Ksum = Obuf + BATCH * CCH * HWSZ;  // 16384
  float* Vsum = Ksum + BATCH * NHEAD * HWSZ; // 16384
  float* Wkh  = Vsum + BATCH * NHEAD * HWSZ; // 512
  float* Wvh  = Wkh + NHEAD * CCH;           // 512
  float* bkh  = Wvh + NHEAD * CCH;           // 8
  float* bvh  = bkh + NHEAD;                 // 8
  float* mm   = bvh + NHEAD;                 // 32

  // 1) Q = Wq @ x1 + bq   (WMMA f32 GEMM)
  gemm64_wmma<<<dim3(HWSZ / 16, CCH / 16, BATCH), 32, 0, stream>>>(
      Wq, bq, x1, x1 /*unused*/, Qbuf, 0);

  // 2) Collapse head_dim out of Wk/Wv (K/V are only ever used summed over n)
  reduce_w<<<2, 256, 0, stream>>>(Wk, bk, Wv, bv, Wkh, bkh, Wvh, bvh);

  // 3) Ksum/Vsum = collapsed 8x64 GEMMs
  kv_sum<<<(BATCH * NHEAD * HWSZ) / 256, 256, 0, stream>>>(
      Wkh, bkh, Wvh, bvh, x2, Ksum, Vsum);

  // 4) Per-(b,h) max/min of Ksum (softmax stabilizer)
  kminmax<<<BATCH * NHEAD, 256, 0, stream>>>(Ksum, mm);

  // 5) Attention core
  attn_kernel<<<dim3(HWSZ / 256, HDIM, BATCH * NHEAD), 256, 0, stream>>>(
      Qbuf, Ksum, Vsum, mm, Obuf);

  // 6) y = Wo @ O + bo + x1  (WMMA f32 GEMM with residual)
  gemm64_wmma<<<dim3(HWSZ / 16, CCH / 16, BATCH), 32, 0, stream>>>(
      Wo, bo, Obuf, x1, out, 1);
}